// DiffIOU_4939212390583
// MI455X (gfx1250) — compile-verified
//
#include <hip/hip_runtime.h>

// DiffIOU for MI455X (gfx1250, wave32).
// One workgroup per batch; per-batch reductions done with chained
// V_WMMA_F32_16X16X4_F32 (A = 6 data rows, B = ones, C accumulates).

typedef __attribute__((ext_vector_type(2))) float v2f;
typedef __attribute__((ext_vector_type(8))) float v8f;

#define DIM 512
#define BS 64
#define NV 128

__global__ __launch_bounds__(NV) void diffiou_batch_kernel(
    const float* __restrict__ pred_pts,   // (BS, NV, 2)
    const float* __restrict__ gt_pts,     // (BS, NV, 2)
    const float* __restrict__ gt_mask,    // (BS, 4, DIM, DIM)
    float* __restrict__ ratios)           // (BS,)
{
  __shared__ float Sx[NV], Sy[NV], Gx[NV], Gy[NV];
  __shared__ float Q[16][NV];   // WMMA A-matrix source rows; rows 6..15 zero

  const int b = blockIdx.x;
  const int v = threadIdx.x;    // 0..127, one vertex per thread

  // ---- load + scale points ----
  const float px = pred_pts[(b * NV + v) * 2 + 0] * (float)DIM;
  const float py = pred_pts[(b * NV + v) * 2 + 1] * (float)DIM;
  const float gx = gt_pts[(b * NV + v) * 2 + 0] * (float)DIM;
  const float gy = gt_pts[(b * NV + v) * 2 + 1] * (float)DIM;
  Sx[v] = px; Sy[v] = py; Gx[v] = gx; Gy[v] = gy;

  // zero unused WMMA rows (6..15)
#pragma unroll
  for (int r = 6; r < 16; ++r) Q[r][v] = 0.0f;
  __syncthreads();

  // ---- per-polygon y-max (redundant scan, tiny) ----
  float ymaxP = -1e30f, ymaxG = -1e30f;
  for (int i = 0; i < NV; ++i) {
    ymaxP = fmaxf(ymaxP, Sy[i]);
    ymaxG = fmaxf(ymaxG, Gy[i]);
  }

  const int nv = (v + 1) & (NV - 1);
  const float px1 = Sx[nv], py1 = Sy[nv];
  const float gx1 = Gx[nv], gy1 = Gy[nv];

  // shoelace-style edge terms
  const float predTerm = (px1 - px) * (ymaxP - (py1 + py) * 0.5f);
  const float gtTerm   = (gx1 - gx) * (ymaxG - (gy1 + gy) * 0.5f);

  // uniqueness masks & signs
  float fx = floorf(px1) - floorf(px);
  float fy = floorf(py1) - floorf(py);
  if (v == 0) { fx = 1.0f; fy = 1.0f; }
  const float xmask = fminf(fabsf(fx), 1.0f);
  const float ymask = fminf(fabsf(fy), 1.0f);
  const float dx = px1 - px, dy = py1 - py;
  const float xsign = (dx > 0.0f) ? 1.0f : ((dx < 0.0f) ? -1.0f : 0.0f);
  const float ysign = (dy > 0.0f) ? 1.0f : ((dy < 0.0f) ? -1.0f : 0.0f);
  const float smx = xsign * xmask;
  const float smy = ysign * ymask;

  // ---- bilinear gather from gt_mask ----
  const float X0f = floorf(px), Y0f = floorf(py);
  const float wx1 = px - X0f, wy1 = py - Y0f;
  const float wx0 = 1.0f - wx1, wy0 = 1.0f - wy1;
  const float w00 = wx0 * wy0, w01 = wx0 * wy1;
  const float w10 = wx1 * wy0, w11 = wx1 * wy1;
  int ix0 = min(max((int)X0f, 0), DIM - 1);
  int iy0 = min(max((int)Y0f, 0), DIM - 1);
  int ix1 = min(max((int)X0f + 1, 0), DIM - 1);
  int iy1 = min(max((int)Y0f + 1, 0), DIM - 1);

#pragma unroll
  for (int c = 0; c < 4; ++c) {
    const float* mb = gt_mask + (size_t)(b * 4 + c) * DIM * DIM;
    const float m00 = mb[iy0 * DIM + ix0];
    const float m01 = mb[iy1 * DIM + ix0];
    const float m10 = mb[iy0 * DIM + ix1];
    const float m11 = mb[iy1 * DIM + ix1];
    const float interp = w00 * m00 + w01 * m01 + w10 * m10 + w11 * m11;
    Q[c][v] = interp * ((c < 2) ? smx : smy);
  }
  Q[4][v] = predTerm;
  Q[5][v] = gtTerm;
  __syncthreads();

  // ---- WMMA reduction: D = A * ones(4x16) + C, chained over 32 K-chunks.
  // A layout (16x4 f32): lanes 0-15 -> K={0,1}, lanes 16-31 -> K={2,3};
  // row M = lane & 15.  All 4 waves compute identical results (uniform EXEC).
  const int lane = threadIdx.x & 31;
  const int m = lane & 15;
  const int kh = (lane >> 4) & 1;
  v8f acc = {};
  v2f ones; ones.x = 1.0f; ones.y = 1.0f;
  for (int j = 0; j < NV / 4; ++j) {
    v2f a;
    a.x = Q[m][4 * j + 2 * kh + 0];
    a.y = Q[m][4 * j + 2 * kh + 1];
    acc = __builtin_amdgcn_wmma_f32_16x16x4_f32(
        /*neg_a=*/false, a, /*neg_b=*/false, ones,
        /*c_mod=*/(short)0, acc, /*reuse_a=*/false, /*reuse_b=*/false);
  }

  // lane 0 holds D[0..7, 0] in acc[0..7]: channel sums in 0..3, areas in 4..5
  if (threadIdx.x == 0) {
    const float int_area = 0.25f * (fabsf(acc[0]) + fabsf(acc[1]) +
                                    fabsf(acc[2]) + fabsf(acc[3]));
    const float pred_area = fabsf(acc[4]);
    const float gt_area   = fabsf(acc[5]);
    const float uni = pred_area + gt_area - int_area;
    ratios[b] = int_area / uni;
  }
}

__global__ void diffiou_mean_kernel(const float* __restrict__ ratios,
                                    float* __restrict__ out)
{
  if (threadIdx.x == 0) {
    float s = 0.0f;
    for (int i = 0; i < BS; ++i) s += ratios[i];
    out[0] = s / (float)BS;
  }
}

extern "C" void kernel_launch(void* const* d_in, const int* in_sizes, int n_in,
                              void* d_out, int out_size, void* d_ws, size_t ws_size,
                              hipStream_t stream) {
  const float* pred = (const float*)d_in[0];  // (64,128,2) f32
  const float* gt   = (const float*)d_in[1];  // (64,128,2) f32
  const float* gm   = (const float*)d_in[2];  // (64,4,512,512) f32
  float* ws  = (float*)d_ws;                  // >= 64 floats
  float* out = (float*)d_out;                 // 1 float

  diffiou_batch_kernel<<<BS, NV, 0, stream>>>(pred, gt, gm, ws);
  diffiou_mean_kernel<<<1, 32, 0, stream>>>(ws, out);
}